// GINConv_42838003810827
// MI455X (gfx1250) — compile-verified
//
#include <hip/hip_runtime.h>
#include <hip/hip_bf16.h>

typedef __attribute__((ext_vector_type(2))) float v2f;
typedef __attribute__((ext_vector_type(8))) float v8f;

#define DIN  32
#define DOUT 32
#define ROWS_PER_WAVE 16
#define WAVES_PER_BLOCK 8
#define ROWS_PER_BLOCK (ROWS_PER_WAVE * WAVES_PER_BLOCK)   // 128
#define LDS_STRIDE 33                                       // pad to avoid bank conflicts

// Fused GIN: X' = A@X (CSR neighbor sum), out = X' @ W.
// Wave-level plan: each wave owns 16 rows.
//   Phase 1: CSR aggregation. rp bounds preloaded once per wave and broadcast
//            via v_readlane; column indices fetched 32-at-a-time with one
//            coalesced 128B load; per-edge gather uses scalar row base +
//            lane*4 (SADDR-form global_load_b32, 128B per edge).
//   Phase 2: 16x v_wmma_f32_16x16x4_f32 for the 16x32 @ 32x32 fp32 GEMM tile.
__global__ __launch_bounds__(256) void gin_fused_kernel(
    const float* __restrict__ X,      // [N, 32]
    const float* __restrict__ W,      // [32, 32]
    const int*   __restrict__ rp,     // [N+1]
    const int*   __restrict__ ci,     // [E]
    float*       __restrict__ out,    // [N, 32]
    int N)
{
    __shared__ float lds[ROWS_PER_BLOCK * LDS_STRIDE];

    const int lane    = threadIdx.x & 31;
    const int wave    = threadIdx.x >> 5;
    const int rowBase = blockIdx.x * ROWS_PER_BLOCK + wave * ROWS_PER_WAVE;

    const int nlane   = lane & 15;   // column within 16-wide tile / A row index
    const int halfSel = lane >> 4;   // 0: low half, 1: high half (K/M offset)

    // ---- per-wave row-pointer preload: lanes 0..16 hold rp[rowBase+lane] ----
    int rpIdx = rowBase + (lane <= 16 ? lane : 16);
    if (rpIdx > N) rpIdx = N;
    const int rowPtr = rp[rpIdx];    // one coalesced load per wave

    // ---------------- Phase 1: CSR neighbor-sum aggregation ----------------
    float* slab = &lds[wave * ROWS_PER_WAVE * LDS_STRIDE];
    for (int rr = 0; rr < ROWS_PER_WAVE; ++rr) {
        const int row = rowBase + rr;
        float acc = 0.0f;
        if (row < N) {                                       // wave-uniform
            const int e0 = __builtin_amdgcn_readlane(rowPtr, rr);
            const int e1 = __builtin_amdgcn_readlane(rowPtr, rr + 1);
            for (int ebase = e0; ebase < e1; ebase += 32) {
                const int cnt = (e1 - ebase) < 32 ? (e1 - ebase) : 32;
                // one coalesced 128B load covers up to 32 edge indices
                const int cols = ci[ebase + (lane < cnt ? lane : cnt - 1)];
                for (int i = 0; i < cnt; ++i) {
                    const int col = __builtin_amdgcn_readlane(cols, i);
                    // scalar row base + lane*4 -> SADDR-form 128B gather
                    acc += X[(long)col * DIN + lane];
                }
            }
        }
        slab[rr * LDS_STRIDE + lane] = acc;
    }
    __syncthreads();   // order LDS stores vs. cross-lane fragment reads

    // ---------------- Phase 2: tile GEMM via V_WMMA_F32_16X16X4_F32 --------
    // B fragments of W (32x32): B is 4x16 per step; VGPR v holds
    // K = k0+v for lanes 0-15 and K = k0+2+v for lanes 16-31.
    v2f bfrag[2][8];
#pragma unroll
    for (int nt = 0; nt < 2; ++nt) {
#pragma unroll
        for (int kk = 0; kk < 8; ++kk) {
            const int k0 = 4 * kk + 2 * halfSel;
            bfrag[nt][kk].x = W[(k0 + 0) * DOUT + nt * 16 + nlane];
            bfrag[nt][kk].y = W[(k0 + 1) * DOUT + nt * 16 + nlane];
        }
    }

    v8f c0 = {};  // N-tile 0 (cols 0..15)
    v8f c1 = {};  // N-tile 1 (cols 16..31)
#pragma unroll
    for (int kk = 0; kk < 8; ++kk) {
        // A fragment (16x4 f32): lane&15 selects row M; lanes 0-15 carry
        // K = 4kk,4kk+1; lanes 16-31 carry K = 4kk+2,4kk+3.
        const int kbase = 4 * kk + 2 * halfSel;
        v2f a;
        a.x = slab[nlane * LDS_STRIDE + kbase + 0];
        a.y = slab[nlane * LDS_STRIDE + kbase + 1];

        c0 = __builtin_amdgcn_wmma_f32_16x16x4_f32(
                false, a, false, bfrag[0][kk], (short)0, c0, false, false);
        c1 = __builtin_amdgcn_wmma_f32_16x16x4_f32(
                false, a, false, bfrag[1][kk], (short)0, c1, false, false);
    }

    // ---------------- Store D (16x16 f32 C/D layout) ------------------------
    // VGPR j: lanes 0-15 -> M=j, N=lane; lanes 16-31 -> M=j+8, N=lane-16.
    if (rowBase + ROWS_PER_WAVE <= N) {
        // fast path: whole 16-row tile in range (all but the last block)
#pragma unroll
        for (int j = 0; j < 8; ++j) {
            const long row = rowBase + j + 8 * halfSel;
            out[row * DOUT +      nlane] = c0[j];
            out[row * DOUT + 16 + nlane] = c1[j];
        }
    } else {
#pragma unroll
        for (int j = 0; j < 8; ++j) {
            const int row = rowBase + j + 8 * halfSel;
            if (row < N) {
                out[(long)row * DOUT +      nlane] = c0[j];
                out[(long)row * DOUT + 16 + nlane] = c1[j];
            }
        }
    }
}

extern "C" void kernel_launch(void* const* d_in, const int* in_sizes, int n_in,
                              void* d_out, int out_size, void* d_ws, size_t ws_size,
                              hipStream_t stream) {
    const float* X  = (const float*)d_in[0];   // [N, 32]
    const float* W  = (const float*)d_in[1];   // [32, 32]
    const int*   rp = (const int*)d_in[2];     // [N+1]
    const int*   ci = (const int*)d_in[3];     // [E]
    float*       out = (float*)d_out;          // [N, 32]

    const int N = in_sizes[2] - 1;             // row_pointers has N+1 entries

    const int blocks = (N + ROWS_PER_BLOCK - 1) / ROWS_PER_BLOCK;
    gin_fused_kernel<<<blocks, 256, 0, stream>>>(X, W, rp, ci, out, N);
}